// HeteroGAT_27522150432789
// MI455X (gfx1250) — compile-verified
//
#include <hip/hip_runtime.h>

typedef float v2f __attribute__((ext_vector_type(2)));
typedef float v8f __attribute__((ext_vector_type(8)));

#define NU 50000
#define NR 100000
#define NI 10000
#define ER 500000
#define EH 300000
#define HID 256
#define CEILDIV(a,b) (((a)+(b)-1)/(b))

// ---------- helpers: order-preserving float<->uint encoding for atomic max ----------
__device__ __forceinline__ unsigned fenc(float f) {
  unsigned u = __float_as_uint(f);
  return (u & 0x80000000u) ? ~u : (u | 0x80000000u);
}
__device__ __forceinline__ float fdec(unsigned u) {
  return __uint_as_float((u & 0x80000000u) ? (u & 0x7FFFFFFFu) : ~u);
}

// ---------- fill ----------
__global__ void fill_f32(float* __restrict__ p, float v, long n) {
  long i = (long)blockIdx.x * blockDim.x + threadIdx.x;
  if (i < n) p[i] = v;
}
__global__ void fill_u32(unsigned* __restrict__ p, unsigned v, long n) {
  long i = (long)blockIdx.x * blockDim.x + threadIdx.x;
  if (i < n) p[i] = v;
}

// ---------- WMMA f32 GEMM: O[N,256] = X[N,K] @ W[K,256] ----------
// One wave computes one 16x16 output tile using V_WMMA_F32_16X16X4_F32.
// A (16x4, 32-bit): lane L holds A[L%16, 2*(L/16)+{0,1}] in 2 VGPRs.
// B (4x16, 32-bit): lane L holds B[2*(L/16)+{0,1}, L%16].
// D (16x16): VGPR v of lane L holds D[v + 8*(L/16), L%16].
// N is always a multiple of 16 here (50000/100000/10000).
__global__ void gemm_wmma(const float* __restrict__ X, const float* __restrict__ W,
                          float* __restrict__ O, int N, int K) {
  const int lane = threadIdx.x & 31;
  const int wave = threadIdx.x >> 5;
  const int tile = blockIdx.x * 8 + wave;
  const int ntiles = (N >> 4) * 16;           // row tiles * 16 col tiles
  if (tile >= ntiles) return;
  const int colTile = tile & 15;
  const int rowTile = tile >> 4;
  const int m  = lane & 15;
  const int kh = lane >> 4;                   // 0 or 1 -> K sub-pair
  const long row = (long)rowTile * 16 + m;
  const int  col = colTile * 16 + m;
  v8f acc = {};
  const float* __restrict__ xrow = X + row * (long)K;
  for (int k0 = 0; k0 < K; k0 += 4) {
    const int ka = k0 + kh * 2;
    v2f a = *(const v2f*)(xrow + ka);         // A[m, ka], A[m, ka+1]
    v2f b;
    b.x = W[(long)ka * HID + col];            // B[ka,   col]
    b.y = W[(long)(ka + 1) * HID + col];      // B[ka+1, col]
    acc = __builtin_amdgcn_wmma_f32_16x16x4_f32(
        false, a, false, b, (short)0, acc, false, false);
  }
  const long obase = ((long)rowTile * 16 + kh * 8) * HID + col;
#pragma unroll
  for (int v = 0; v < 8; ++v) O[obase + (long)v * HID] = acc[v];
}

// ---------- fold attention vector into weights: V[k,h] = sum_c W[k,h*64+c]*a[h,c] ----------
__global__ void fold_attn(const float* __restrict__ W, const float* __restrict__ av,
                          float* __restrict__ V, int K) {
  int idx = blockIdx.x * blockDim.x + threadIdx.x;
  if (idx >= K * 4) return;
  int k = idx >> 2, h = idx & 3;
  float s = 0.f;
  const float* wr = W + (long)k * HID + h * 64;
  const float* ar = av + h * 64;
  for (int c = 0; c < 64; ++c) s += wr[c] * ar[c];
  V[k * 4 + h] = s;
}

// ---------- per-node attention logits: A[n,h] = sum_k X[n,k] * V[k,h] ----------
__global__ void attn_proj(const float* __restrict__ X, const float* __restrict__ V,
                          float* __restrict__ A, int N, int K) {
  int n = blockIdx.x * blockDim.x + threadIdx.x;
  if (n >= N) return;
  float s0 = 0.f, s1 = 0.f, s2 = 0.f, s3 = 0.f;
  const float* xr = X + (long)n * K;
  for (int k = 0; k < K; ++k) {
    float x = xr[k];
    s0 += x * V[k * 4 + 0];
    s1 += x * V[k * 4 + 1];
    s2 += x * V[k * 4 + 2];
    s3 += x * V[k * 4 + 3];
  }
  A[n * 4 + 0] = s0; A[n * 4 + 1] = s1; A[n * 4 + 2] = s2; A[n * 4 + 3] = s3;
}

// ---------- edge pass A: leaky-relu logits + segment max (via uint-encoded atomicMax) ----------
__global__ void edge_logits(const int* __restrict__ es, const int* __restrict__ ed,
                            const float* __restrict__ As, const float* __restrict__ Ad,
                            float* __restrict__ el, unsigned* __restrict__ mbuf, int E) {
  int idx = blockIdx.x * blockDim.x + threadIdx.x;
  if (idx >= E * 4) return;
  int e = idx >> 2, h = idx & 3;
  int s = es[e], d = ed[e];
  float v = As[(long)s * 4 + h] + Ad[(long)d * 4 + h];
  v = (v > 0.f) ? v : 0.2f * v;                       // leaky_relu 0.2
  el[idx] = v;
  atomicMax(&mbuf[(long)d * 4 + h], fenc(v));
}

// ---------- edge pass B: ex = exp(el - max); denom += ex ----------
__global__ void edge_exp(const int* __restrict__ ed, float* __restrict__ el,
                         const unsigned* __restrict__ mbuf, float* __restrict__ denom, int E) {
  int idx = blockIdx.x * blockDim.x + threadIdx.x;
  if (idx >= E * 4) return;
  int e = idx >> 2, h = idx & 3;
  int d = ed[e];
  float ex = expf(el[idx] - fdec(mbuf[(long)d * 4 + h]));
  el[idx] = ex;
  atomicAdd(&denom[(long)d * 4 + h], ex);
}

// ---------- edge pass C: acc[dst,:] += alpha[h] * hs[src,:]  (one wave per edge) ----------
__global__ void edge_scatter(const int* __restrict__ es, const int* __restrict__ ed,
                             const float* __restrict__ ex, const float* __restrict__ denom,
                             const float* __restrict__ hs, float* __restrict__ acc, int E) {
  int wave = threadIdx.x >> 5;
  int lane = threadIdx.x & 31;
  long e = (long)blockIdx.x * 8 + wave;
  if (e >= E) return;
  int s = es[e], d = ed[e];
  int h = lane >> 3;                                   // 8 lanes per head
  float alpha = ex[e * 4 + h] / (denom[(long)d * 4 + h] + 1e-16f);
  const float4* hv = (const float4*)(hs + (long)s * HID + lane * 8);
  float4 v0 = hv[0], v1 = hv[1];
  float* out = acc + (long)d * HID + lane * 8;
  atomicAdd(out + 0, v0.x * alpha);
  atomicAdd(out + 1, v0.y * alpha);
  atomicAdd(out + 2, v0.z * alpha);
  atomicAdd(out + 3, v0.w * alpha);
  atomicAdd(out + 4, v1.x * alpha);
  atomicAdd(out + 5, v1.y * alpha);
  atomicAdd(out + 6, v1.z * alpha);
  atomicAdd(out + 7, v1.w * alpha);
}

// ---------- finalize: out = maybe_combine(prev, relu?((acc + b0 [+ b1]) * pre * w)) ----------
__global__ void finalize_k(const float* __restrict__ acc, const float* __restrict__ b0,
                           const float* __restrict__ b1, const float* __restrict__ wa,
                           const float* __restrict__ wb, float pre,
                           const float* __restrict__ prev, float* __restrict__ out,
                           long total, int do_relu) {
  long i = (long)blockIdx.x * blockDim.x + threadIdx.x;
  if (i >= total) return;
  int c = (int)(i & (HID - 1));
  float b = b0[c];
  if (b1) b += b1[c];
  float w = wa[0];
  if (wb) w = 0.5f * (w + wb[0]);
  float v = (acc[i] + b) * pre * w;
  if (do_relu) v = fmaxf(v, 0.f);
  if (prev) v = 0.5f * (prev[i] + v);
  out[i] = v;
}

extern "C" void kernel_launch(void* const* d_in, const int* in_sizes, int n_in,
                              void* d_out, int out_size, void* d_ws, size_t ws_size,
                              hipStream_t stream) {
  (void)in_sizes; (void)n_in; (void)out_size; (void)ws_size;
  const float* x_user   = (const float*)d_in[0];
  const float* x_recipe = (const float*)d_in[1];
  const float* x_ing    = (const float*)d_in[2];
  const int* ers = (const int*)d_in[3];
  const int* erd = (const int*)d_in[4];
  const int* ehs = (const int*)d_in[5];
  const int* ehd = (const int*)d_in[6];
  auto P = [&](int i) { return (const float*)d_in[i]; };
  // params flattened in setup_inputs() insertion order:
  // l1: rates@7, rev_rates@12, has@17, rev_has@22 ; l2: rates@27, rev_rates@32, has@37, rev_has@42
  // each rel: +0 Ws, +1 Wd, +2 as, +3 ad, +4 b ; then w_rates@47, w_rev_rates@48, w_has@49, w_rev_has@50
  const float* w_rates     = P(47);
  const float* w_rev_rates = P(48);
  const float* w_has       = P(49);
  const float* w_rev_has   = P(50);

  float* wsp = (float*)d_ws;
  size_t off = 0;
  auto alloc = [&](size_t n) { float* p = wsp + off; off += n; return p; };
  float* feat_u = alloc((size_t)NU * HID);
  float* feat_r = alloc((size_t)NR * HID);
  float* feat_i = alloc((size_t)NI * HID);
  float* acc_u  = alloc((size_t)NU * HID);
  float* acc_r  = alloc((size_t)NR * HID);
  float* acc_i  = alloc((size_t)NI * HID);
  float* hsbuf  = alloc((size_t)NR * HID);       // max Ns = NR
  float* a_s    = alloc((size_t)NR * 4);
  float* a_d    = alloc((size_t)NR * 4);
  unsigned* mbuf = (unsigned*)alloc((size_t)NR * 4);
  float* denom  = alloc((size_t)NR * 4);
  float* elbuf  = alloc((size_t)ER * 4);         // max E
  float* Vs     = alloc(HID * 4);
  float* Vd     = alloc(HID * 4);

  const int B = 256;

  auto run_rel = [&](const float* xs, int Ns, const float* xd, int Nd,
                     const int* es, const int* ed, int E,
                     int pbase, int K, float* acc) {
    const float* Ws  = P(pbase + 0);
    const float* Wd  = P(pbase + 1);
    const float* avs = P(pbase + 2);
    const float* avd = P(pbase + 3);
    fold_attn<<<CEILDIV(K * 4, B), B, 0, stream>>>(Ws, avs, Vs, K);
    fold_attn<<<CEILDIV(K * 4, B), B, 0, stream>>>(Wd, avd, Vd, K);
    int ntiles = (Ns >> 4) * 16;
    gemm_wmma<<<CEILDIV(ntiles, 8), B, 0, stream>>>(xs, Ws, hsbuf, Ns, K);
    attn_proj<<<CEILDIV(Ns, B), B, 0, stream>>>(xs, Vs, a_s, Ns, K);
    attn_proj<<<CEILDIV(Nd, B), B, 0, stream>>>(xd, Vd, a_d, Nd, K);
    fill_u32<<<CEILDIV(Nd * 4, B), B, 0, stream>>>(mbuf, 0u, (long)Nd * 4);
    fill_f32<<<CEILDIV(Nd * 4, B), B, 0, stream>>>(denom, 0.f, (long)Nd * 4);
    edge_logits<<<CEILDIV(E * 4, B), B, 0, stream>>>(es, ed, a_s, a_d, elbuf, mbuf, E);
    edge_exp<<<CEILDIV(E * 4, B), B, 0, stream>>>(ed, elbuf, mbuf, denom, E);
    edge_scatter<<<CEILDIV(E, 8), B, 0, stream>>>(es, ed, elbuf, denom, hsbuf, acc, E);
  };

  auto zero_accs = [&]() {
    fill_f32<<<CEILDIV((long)NU * HID, B), B, 0, stream>>>(acc_u, 0.f, (long)NU * HID);
    fill_f32<<<CEILDIV((long)NR * HID, B), B, 0, stream>>>(acc_r, 0.f, (long)NR * HID);
    fill_f32<<<CEILDIV((long)NI * HID, B), B, 0, stream>>>(acc_i, 0.f, (long)NI * HID);
  };

  // ===== Layer 1 (K = 64) =====
  zero_accs();
  run_rel(x_user,   NU, x_recipe, NR, ers, erd, ER,  7, 64, acc_r);  // rates: user -> recipe
  run_rel(x_recipe, NR, x_user,   NU, erd, ers, ER, 12, 64, acc_u);  // rev_rates: recipe -> user
  run_rel(x_recipe, NR, x_ing,    NI, ehs, ehd, EH, 17, 64, acc_i);  // has: recipe -> ing
  run_rel(x_ing,    NI, x_recipe, NR, ehd, ehs, EH, 22, 64, acc_r);  // rev_has: ing -> recipe
  finalize_k<<<CEILDIV((long)NU * HID, B), B, 0, stream>>>(
      acc_u, P(16), nullptr, w_rev_rates, nullptr, 1.f, nullptr, feat_u, (long)NU * HID, 1);
  finalize_k<<<CEILDIV((long)NR * HID, B), B, 0, stream>>>(
      acc_r, P(11), P(26), w_rates, w_has, 0.5f, nullptr, feat_r, (long)NR * HID, 1);
  finalize_k<<<CEILDIV((long)NI * HID, B), B, 0, stream>>>(
      acc_i, P(21), nullptr, w_rev_has, nullptr, 1.f, nullptr, feat_i, (long)NI * HID, 1);

  // ===== Layer 2 (K = 256) =====
  zero_accs();
  run_rel(feat_u, NU, feat_r, NR, ers, erd, ER, 27, 256, acc_r);
  run_rel(feat_r, NR, feat_u, NU, erd, ers, ER, 32, 256, acc_u);
  run_rel(feat_r, NR, feat_i, NI, ehs, ehd, EH, 37, 256, acc_i);
  run_rel(feat_i, NI, feat_r, NR, ehd, ehs, EH, 42, 256, acc_r);

  float* out_u = (float*)d_out;
  float* out_r = out_u + (size_t)NU * HID;
  float* out_i = out_r + (size_t)NR * HID;
  finalize_k<<<CEILDIV((long)NU * HID, B), B, 0, stream>>>(
      acc_u, P(36), nullptr, w_rev_rates, nullptr, 1.f, feat_u, out_u, (long)NU * HID, 0);
  finalize_k<<<CEILDIV((long)NR * HID, B), B, 0, stream>>>(
      acc_r, P(31), P(46), w_rates, w_has, 0.5f, feat_r, out_r, (long)NR * HID, 0);
  finalize_k<<<CEILDIV((long)NI * HID, B), B, 0, stream>>>(
      acc_i, P(41), nullptr, w_rev_has, nullptr, 1.f, feat_i, out_i, (long)NI * HID, 0);
}